// SwitchFeedForward_18476949307737
// MI455X (gfx1250) — compile-verified
//
#include <hip/hip_runtime.h>

// ---------------------------------------------------------------------------
// Switch (MoE) feed-forward for MI455X / gfx1250, fp32 via V_WMMA_F32_16X16X4_F32
// Tokens are routed (top-1), compacted per expert, then processed in 16-token
// tiles with wave32 WMMA GEMMs. Only ~17 GFLOP of routed work instead of the
// reference's 137 GFLOP dense-all-experts formulation.
// ---------------------------------------------------------------------------

typedef float v2f __attribute__((ext_vector_type(2)));
typedef float v8f __attribute__((ext_vector_type(8)));

#define TOK 8192      // SEQ * BATCH
#define DM  512       // d_model
#define DF  1024      // d_ff
#define NE  8         // experts

// Output layout (flattened tuple, fp32):
//   [0, TOK*DM)            : out
//   [CNT_OFF, +8)          : counts (float)
//   [SUM_OFF, +8)          : route_prob.sum(0)
//   [ZERO_OFF]             : literal 0
//   [PMAX_OFF, +TOK)       : route_prob_max
#define CNT_OFF  (TOK * DM)
#define SUM_OFF  (CNT_OFF + NE)
#define ZERO_OFF (SUM_OFF + NE)
#define PMAX_OFF (ZERO_OFF + 1)

// Workspace layout (as int32 slots; probs region reinterpreted as float)
#define WS_ROUTE 0                    // [TOK] expert id per token
#define WS_CNT   (TOK)                // [8]   tokens per expert (int)
#define WS_OFF   (TOK + 8)            // [9]   exclusive prefix
#define WS_FILL  (TOK + 20)           // [8]   scatter cursors
#define WS_LIST  (TOK + 32)           // [TOK] token ids grouped by expert
#define WS_PROBS (TOK + 32 + TOK)     // [TOK*8] floats: per-token softmax probs

// ---------------------------------------------------------------------------
__global__ void k_init(int* __restrict__ wsI) {
    int t = threadIdx.x;
    if (t < NE) wsI[WS_CNT + t] = 0;
}

// ---------------------------------------------------------------------------
// Router: one wave per token. Each lane covers 16 contiguous d values.
__global__ __launch_bounds__(256)
void k_router(const float* __restrict__ x, const float* __restrict__ Wsw,
              const float* __restrict__ bsw, float* __restrict__ outp,
              int* __restrict__ wsI, float* __restrict__ probs) {
    int lane = threadIdx.x & 31;
    int wave = threadIdx.x >> 5;
    int t = blockIdx.x * 8 + wave;
    if (t >= TOK) return;

    const float* xr = x + (size_t)t * DM + lane * 16;
    const float* Wr = Wsw + lane * 16 * NE;

    float acc[NE];
#pragma unroll
    for (int e = 0; e < NE; e++) acc[e] = 0.f;
#pragma unroll
    for (int i = 0; i < 16; i++) {
        float xv = xr[i];
#pragma unroll
        for (int e = 0; e < NE; e++) acc[e] = fmaf(xv, Wr[i * NE + e], acc[e]);
    }
    // wave32 butterfly reduce -> every lane holds full dot products
#pragma unroll
    for (int off = 16; off >= 1; off >>= 1) {
#pragma unroll
        for (int e = 0; e < NE; e++) acc[e] += __shfl_xor(acc[e], off, 32);
    }

    if (lane == 0) {
        float l[NE];
#pragma unroll
        for (int e = 0; e < NE; e++) l[e] = acc[e] + bsw[e];
        // argmax (first occurrence, matches jnp.argmax)
        int am = 0; float bm = l[0];
#pragma unroll
        for (int e = 1; e < NE; e++) if (l[e] > bm) { bm = l[e]; am = e; }
        // softmax
        float s = 0.f, p[NE];
#pragma unroll
        for (int e = 0; e < NE; e++) { p[e] = __expf(l[e] - bm); s += p[e]; }
        float inv = 1.f / s;
        float pmax = 0.f;
#pragma unroll
        for (int e = 0; e < NE; e++) { p[e] *= inv; pmax = fmaxf(pmax, p[e]); }

        wsI[WS_ROUTE + t] = am;
        atomicAdd(&wsI[WS_CNT + am], 1);          // exact int count
        outp[PMAX_OFF + t] = pmax;
#pragma unroll
        for (int e = 0; e < NE; e++) probs[t * NE + e] = p[e];
    }
}

// ---------------------------------------------------------------------------
__global__ void k_offsets(int* __restrict__ wsI, float* __restrict__ outp) {
    int t = threadIdx.x;
    if (t == 0) {
        int a = 0;
        for (int e = 0; e < NE; e++) { wsI[WS_OFF + e] = a; a += wsI[WS_CNT + e]; }
        wsI[WS_OFF + NE] = a;
        outp[ZERO_OFF] = 0.f;
    }
    if (t < NE) {
        wsI[WS_FILL + t] = 0;
        outp[CNT_OFF + t] = (float)wsI[WS_CNT + t];
    }
}

// Deterministic (fixed-order) reduction of route_prob over tokens.
__global__ __launch_bounds__(256)
void k_probsum(const float* __restrict__ probs, float* __restrict__ outp) {
    __shared__ float red[256];
    int e = blockIdx.x, tid = threadIdx.x;
    float s = 0.f;
    for (int t = tid; t < TOK; t += 256) s += probs[t * NE + e];
    red[tid] = s;
    __syncthreads();
    for (int st = 128; st > 0; st >>= 1) {
        if (tid < st) red[tid] += red[tid + st];
        __syncthreads();
    }
    if (tid == 0) outp[SUM_OFF + e] = red[0];
}

// ---------------------------------------------------------------------------
__global__ __launch_bounds__(256)
void k_scatter(int* __restrict__ wsI) {
    int t = blockIdx.x * 256 + threadIdx.x;
    if (t >= TOK) return;
    int e = wsI[WS_ROUTE + t];
    int pos = wsI[WS_OFF + e] + atomicAdd(&wsI[WS_FILL + e], 1);
    wsI[WS_LIST + pos] = t;
}

// ---------------------------------------------------------------------------
// Expert FFN: one block = one 16-token tile of one expert. 8 waves.
// GEMM1: [16 x 512] @ [512 x 1024] -> relu -> LDS h (64 KB)
// GEMM2: [16 x 1024] @ [1024 x 512] -> *p_max -> scatter to out
__global__ __launch_bounds__(256)
void k_ffn(const float* __restrict__ x,
           const float* __restrict__ W1, const float* __restrict__ b1,
           const float* __restrict__ W2, const float* __restrict__ b2,
           float* __restrict__ outp, const int* __restrict__ wsI) {
    __shared__ float hs[16 * DF];   // 64 KB of the 320 KB WGP LDS

    int tid = threadIdx.x;
    int lane = tid & 31;
    int wave = tid >> 5;
    int lm = lane & 15;      // row/col-in-tile index
    int half = lane >> 4;    // K-pair selector per ISA 16x4 A layout

    // Map blockIdx -> (expert, tile) by scanning the 8 counts (uniform).
    int bid = blockIdx.x;
    int e = -1, m = 0, cnt_e = 0, acc = 0;
#pragma unroll
    for (int i = 0; i < NE; i++) {
        int c = wsI[WS_CNT + i];
        int nt = (c + 15) >> 4;
        if (e < 0 && bid < acc + nt) { e = i; m = bid - acc; cnt_e = c; }
        acc += nt;
    }
    if (e < 0) return;   // uniform over the block: EXEC stays all-ones below

    int base = wsI[WS_OFF + e] + m * 16;
    int row = m * 16 + lm;
    int tk = (row < cnt_e) ? wsI[WS_LIST + base + lm] : -1;
    float pmv = (tk >= 0) ? outp[PMAX_OFF + tk] : 0.f;
    // Clamped row pointer: always safe to load; padded rows are zeroed via mask
    // multiply (branchless -> single global_load_b64, no exec-mask juggling).
    const float* xrow = x + (size_t)(tk >= 0 ? tk : 0) * DM;
    float mk = (tk >= 0) ? 1.0f : 0.0f;

    const float* W1e = W1 + (size_t)e * DM * DF;
    const float* b1e = b1 + (size_t)e * DF;
    const float* W2e = W2 + (size_t)e * DF * DM;
    const float* b2e = b2 + (size_t)e * DM;

    v8f zero = {};

    // ---- GEMM1: this wave owns 8 N-tiles (128 columns) of d_ff ----
    v8f c1[8];
#pragma unroll
    for (int i = 0; i < 8; i++) c1[i] = zero;

    int n0base = wave * 128;
    for (int k = 0; k < DM; k += 4) {
        int kk = k + 2 * half;                    // kk is even -> 8B aligned
        v2f a = *(const v2f*)(xrow + kk);         // A: M=lm, K=kk..kk+1
        a.x *= mk;
        a.y *= mk;
        const float* Bp = W1e + (size_t)kk * DF + n0base + lm;
#pragma unroll
        for (int nt = 0; nt < 8; nt++) {
            v2f b;
            b.x = Bp[nt * 16];        // B: K=kk,   N=n0+lm
            b.y = Bp[DF + nt * 16];   //    K=kk+1, N=n0+lm
            c1[nt] = __builtin_amdgcn_wmma_f32_16x16x4_f32(
                false, a, false, b, (short)0, c1[nt], false, false);
        }
    }
    // bias + relu -> LDS   (C layout: vgpr r, lanes0-15 M=r, lanes16-31 M=r+8)
#pragma unroll
    for (int nt = 0; nt < 8; nt++) {
        int n = n0base + nt * 16 + lm;
        float bias = b1e[n];
#pragma unroll
        for (int r = 0; r < 8; r++) {
            int M = r + 8 * half;
            hs[M * DF + n] = fmaxf(c1[nt][r] + bias, 0.f);
        }
    }
    __syncthreads();

    // ---- GEMM2: this wave owns 4 N-tiles (64 columns) of d_model ----
    v8f c2[4];
#pragma unroll
    for (int i = 0; i < 4; i++) c2[i] = zero;

    int n0b = wave * 64;
    for (int k = 0; k < DF; k += 4) {
        int kk = k + 2 * half;                          // even -> 8B aligned
        v2f a = *(const v2f*)(&hs[lm * DF + kk]);       // one ds_load_b64
        const float* Bp = W2e + (size_t)kk * DM + n0b + lm;
#pragma unroll
        for (int nt = 0; nt < 4; nt++) {
            v2f b;
            b.x = Bp[nt * 16];
            b.y = Bp[DM + nt * 16];
            c2[nt] = __builtin_amdgcn_wmma_f32_16x16x4_f32(
                false, a, false, b, (short)0, c2[nt], false, false);
        }
    }
    // bias, scale by routing prob, scatter rows back to token order
#pragma unroll
    for (int nt = 0; nt < 4; nt++) {
        int n = n0b + nt * 16 + lm;
        float bias = b2e[n];
#pragma unroll
        for (int r = 0; r < 8; r++) {
            int M = r + 8 * half;
            int   tkM = __shfl(tk, M, 32);
            float pmM = __shfl(pmv, M, 32);
            if (tkM >= 0)
                outp[(size_t)tkM * DM + n] = (c2[nt][r] + bias) * pmM;
        }
    }
}

// ---------------------------------------------------------------------------
extern "C" void kernel_launch(void* const* d_in, const int* in_sizes, int n_in,
                              void* d_out, int out_size, void* d_ws, size_t ws_size,
                              hipStream_t stream) {
    (void)in_sizes; (void)n_in; (void)out_size; (void)ws_size;

    const float* x   = (const float*)d_in[0];
    const float* Wsw = (const float*)d_in[1];
    const float* bsw = (const float*)d_in[2];
    const float* W1  = (const float*)d_in[3];
    const float* b1  = (const float*)d_in[4];
    const float* W2  = (const float*)d_in[5];
    const float* b2  = (const float*)d_in[6];
    float* outp = (float*)d_out;
    int*   wsI  = (int*)d_ws;
    float* probs = (float*)(wsI + WS_PROBS);

    k_init   <<<1,        32, 0, stream>>>(wsI);
    k_router <<<TOK / 8, 256, 0, stream>>>(x, Wsw, bsw, outp, wsI, probs);
    k_offsets<<<1,        32, 0, stream>>>(wsI, outp);
    k_probsum<<<NE,      256, 0, stream>>>(probs, outp);
    k_scatter<<<TOK/256, 256, 0, stream>>>(wsI);
    // max tiles = ceil(TOK/16) + (NE-1) padding tiles = 519; launch 520
    k_ffn    <<<520,     256, 0, stream>>>(x, W1, b1, W2, b2, outp, wsI);
}